// FocalLoss_68161130988174
// MI455X (gfx1250) — compile-verified
//
#include <hip/hip_runtime.h>
#include <math.h>

// ---------------- problem constants ----------------
#define BATCH   128
#define NBOX    8732
#define NCLS    21
#define BN      (BATCH * NBOX)
#define ALPHA_C 0.25f
#define THREADS 256
#define NBLK0   512      // blocks for elementwise pass (multiple of 64)
#define NBLK3   512      // blocks for masked-CE pass (multiple of 64)

typedef __attribute__((ext_vector_type(2))) float v2f;
typedef __attribute__((ext_vector_type(8))) float v8f;

// ---------------- helpers ----------------
__device__ __forceinline__ float sl1(float d) {
    float ad = fabsf(d);
    return ad < 1.0f ? 0.5f * d * d : ad - 0.5f;
}

// monotone float -> uint map (handles negatives; descending rank == key descending)
__device__ __forceinline__ unsigned f2ord(float f) {
    unsigned u = __float_as_uint(f);
    return (u & 0x80000000u) ? ~u : (u | 0x80000000u);
}

// deterministic fixed-tree block reduction (256 threads)
__device__ __forceinline__ float blk_reduce(float v, float* red) {
    const int tid = threadIdx.x;
    red[tid] = v;
    __syncthreads();
#pragma unroll
    for (int s = THREADS / 2; s > 0; s >>= 1) {
        if (tid < s) red[tid] += red[tid + s];
        __syncthreads();
    }
    float r = red[0];
    __syncthreads();
    return r;
}

// Deterministic exact sum of count64*64 f32 values via chained
// V_WMMA_F32_16X16X4_F32 with an all-ones B matrix (D[m][n] = sum_k A[m][k] + C).
// fp32 multiply by 1.0 is exact; accumulation order fixed by HW -> deterministic.
// Requires a full wave (32 lanes, EXEC all ones).
__device__ __forceinline__ float wmma_sum(const float* __restrict__ src,
                                          int count64, int lane) {
    const int m  = lane & 15;
    const int kk = (lane >> 4) << 1;   // lanes 0-15: K=0,1 ; lanes 16-31: K=2,3
    v2f ones; ones.x = 1.0f; ones.y = 1.0f;
    v8f acc = {};
    for (int c = 0; c < count64; ++c) {
        const float* s = src + c * 64;
        v2f a;
        a.x = s[m * 4 + kk];
        a.y = s[m * 4 + kk + 1];
        acc = __builtin_amdgcn_wmma_f32_16x16x4_f32(
            false, a, false, ones, (short)0, acc, false, false);
    }
    // lane 0 holds rowsums M=0..7 in acc[0..7]; lane 16 holds M=8..15
    float t = acc[0] + acc[1] + acc[2] + acc[3] + acc[4] + acc[5] + acc[6] + acc[7];
    return __shfl(t, 0) + __shfl(t, 16);
}

// ---------------- K0: elementwise pass ----------------
// per box: focal loss term, smooth-L1 terms, CE + key + flags; 6 global partial sums
__global__ __launch_bounds__(THREADS) void k0_elem(
    const float* __restrict__ loc1, const float* __restrict__ loc2,
    const float* __restrict__ loct, const float* __restrict__ conf,
    const int* __restrict__ ctg, const float* __restrict__ osp,
    const int* __restrict__ ost,
    float* __restrict__ ce_buf, unsigned* __restrict__ key_buf,
    unsigned char* __restrict__ flags, float* __restrict__ part) {
    __shared__ float red[THREADS];
    float acc_os = 0.f, acc_l1 = 0.f, acc_l2 = 0.f;
    float cnt_pos = 0.f, cnt_pos_os = 0.f, cnt_osp = 0.f;

    for (int idx = blockIdx.x * THREADS + threadIdx.x; idx < BN;
         idx += NBLK0 * THREADS) {
        const int ct   = ctg[idx];
        const bool pos = ct > 0;
        const bool ign = ct < 0;

        // ---- objectness focal ----
        const float2 o = ((const float2*)osp)[idx];
        const int ot   = ost[idx];
        float m2    = fmaxf(o.x, o.y);
        float lse2  = m2 + logf(expf(o.x - m2) + expf(o.y - m2));
        int   y     = (ot > 0) ? 1 : 0;
        float logpt = (y ? o.y : o.x) - lse2;
        float pt    = expf(logpt);
        float at    = y ? ALPHA_C : (1.0f - ALPHA_C);
        float fl    = -at * logpt * (1.0f - pt) * (1.0f - pt);
        if (ot > -1) acc_os += fl;
        if (ot > 0)  cnt_pos_os += 1.0f;
        bool ospos = expf(o.y - lse2) > 0.4f;   // softmax[...,1] > 0.4
        if (ospos) cnt_osp += 1.0f;

        // ---- localization smooth-L1 (positives only) ----
        if (pos) {
            cnt_pos += 1.0f;
            float4 p1 = ((const float4*)loc1)[idx];
            float4 p2 = ((const float4*)loc2)[idx];
            float4 t  = ((const float4*)loct)[idx];
            acc_l1 += sl1(p1.x - t.x) + sl1(p1.y - t.y) +
                      sl1(p1.z - t.z) + sl1(p1.w - t.w);
            acc_l2 += sl1(p2.x - t.x) + sl1(p2.y - t.y) +
                      sl1(p2.z - t.z) + sl1(p2.w - t.w);
        }

        // ---- 21-class cross entropy ----
        const float* cp = conf + (size_t)idx * NCLS;
        __builtin_prefetch(cp + (size_t)NBLK0 * THREADS * NCLS, 0, 1);
        float mx = cp[0];
#pragma unroll
        for (int c = 1; c < NCLS; ++c) mx = fmaxf(mx, cp[c]);
        float s = 0.0f;
#pragma unroll
        for (int c = 0; c < NCLS; ++c) s += expf(cp[c] - mx);
        float lse = mx + logf(s);
        int   tg  = ct < 0 ? 0 : (ct > NCLS - 1 ? NCLS - 1 : ct);
        float ce  = lse - cp[tg];
        ce_buf[idx]  = ce;
        float cl     = (pos || ign) ? 0.0f : ce;   // stop-grad mining score
        key_buf[idx] = f2ord(cl);
        flags[idx]   = (unsigned char)((pos ? 1 : 0) | (ign ? 2 : 0) | (ospos ? 4 : 0));
    }

    float t0 = blk_reduce(acc_os,     red);
    float t1 = blk_reduce(cnt_pos_os, red);
    float t2 = blk_reduce(cnt_pos,    red);
    float t3 = blk_reduce(acc_l1,     red);
    float t4 = blk_reduce(acc_l2,     red);
    float t5 = blk_reduce(cnt_osp,    red);
    if (threadIdx.x == 0) {
        part[0 * NBLK0 + blockIdx.x] = t0;
        part[1 * NBLK0 + blockIdx.x] = t1;
        part[2 * NBLK0 + blockIdx.x] = t2;
        part[3 * NBLK0 + blockIdx.x] = t3;
        part[4 * NBLK0 + blockIdx.x] = t4;
        part[5 * NBLK0 + blockIdx.x] = t5;
    }
}

// ---------------- K1: per-row hard-negative mining ----------------
// Exact equivalent of rank = argsort(argsort(-cl)) ; neg = rank < num_neg:
// 32-bit radix-select of the R-th largest key, then stable index-order
// tie-break among keys equal to the threshold.
__global__ __launch_bounds__(THREADS) void k1_mine(
    const unsigned* __restrict__ key_buf, unsigned char* __restrict__ flags) {
    __shared__ unsigned sh_keys[NBOX];
    __shared__ unsigned sh_cnt;
    __shared__ unsigned sh_wave[THREADS / 32];
    __shared__ unsigned sh_running;

    const int b   = blockIdx.x;
    const int tid = threadIdx.x;
    const unsigned* krow = key_buf + (size_t)b * NBOX;
    unsigned char*  frow = flags + (size_t)b * NBOX;

    if (tid == 0) sh_cnt = 0u;
    __syncthreads();
    unsigned lp = 0;
    for (int n = tid; n < NBOX; n += THREADS) {
        sh_keys[n] = krow[n];
        lp += (unsigned)(frow[n] & 1);         // pos count for this row
    }
    atomicAdd(&sh_cnt, lp);
    __syncthreads();
    int num_pos = (int)sh_cnt;
    int Ri = 3 * num_pos;
    if (Ri < 1) Ri = 1;
    if (Ri > NBOX - 1) Ri = NBOX - 1;
    const unsigned R = (unsigned)Ri;
    __syncthreads();

    // radix-select: max T with count(key >= T) >= R  ==  R-th largest key
    unsigned T = 0u;
    for (int bit = 31; bit >= 0; --bit) {
        unsigned trial = T | (1u << bit);
        unsigned pref  = trial >> bit;
        if (tid == 0) sh_cnt = 0u;
        __syncthreads();
        unsigned c = 0;
        for (int n = tid; n < NBOX; n += THREADS)
            c += ((sh_keys[n] >> bit) >= pref) ? 1u : 0u;
        atomicAdd(&sh_cnt, c);
        __syncthreads();
        if (sh_cnt >= R) T = trial;            // uniform across block
        __syncthreads();
    }

    // G = strictly-greater count
    if (tid == 0) sh_cnt = 0u;
    __syncthreads();
    {
        unsigned c = 0;
        for (int n = tid; n < NBOX; n += THREADS)
            c += (sh_keys[n] > T) ? 1u : 0u;
        atomicAdd(&sh_cnt, c);
    }
    __syncthreads();
    const unsigned G    = sh_cnt;
    const unsigned Rrem = R - G;               // >= 1 by construction

    // ordered (index-stable) prefix over key==T -> first Rrem ties are neg
    if (tid == 0) sh_running = 0u;
    __syncthreads();
    const int lane = tid & 31;
    const int wv   = tid >> 5;
    for (int base = 0; base < NBOX; base += THREADS) {
        int  n   = base + tid;
        bool inb = n < NBOX;
        unsigned key = inb ? sh_keys[n] : 0u;
        bool gt = inb && (key > T);
        bool eq = inb && (key == T);
        unsigned long long bal = __ballot(eq ? 1 : 0);
        unsigned pre  = (unsigned)__popcll(bal & ((1ull << lane) - 1ull));
        unsigned wtot = (unsigned)__popcll(bal);
        if (lane == 0) sh_wave[wv] = wtot;
        __syncthreads();
        unsigned off = sh_running;
        for (int w = 0; w < wv; ++w) off += sh_wave[w];
        bool neg = gt || (eq && (off + pre) < Rrem);
        if (inb && neg) frow[n] = (unsigned char)(frow[n] | 8);
        __syncthreads();
        if (tid == 0) {
            unsigned tot = 0;
            for (int w = 0; w < THREADS / 32; ++w) tot += sh_wave[w];
            sh_running += tot;
        }
        __syncthreads();
    }
}

// ---------------- K2: global scalars (one wave, WMMA reductions) ----------------
__global__ __launch_bounds__(32) void k2_scalars(
    const float* __restrict__ part, float* __restrict__ scal,
    float* __restrict__ out) {
    const int lane = threadIdx.x;
    float tot[6];
#pragma unroll
    for (int q = 0; q < 6; ++q)
        tot[q] = wmma_sum(part + q * NBLK0, NBLK0 / 64, lane);

    float os_sum  = tot[0] * 10.0f;
    float pos_os  = tot[1];
    float reg_num = tot[2];
    float os_loss = (pos_os > 0.0f) ? (os_sum / fmaxf(pos_os, 1.0f))
                                    : (os_sum / 500.0f);
    float loc_loss = (0.5f * tot[4] + 0.35f * tot[3]) / fmaxf(reg_num, 1.0f);
    if (lane == 0) {
        out[0]  = (reg_num > 0.0f) ? loc_loss : 0.0f;
        out[1]  = os_loss;
        scal[0] = (tot[5] > reg_num * 10.0f) ? 1.0f : 0.0f;  // cond
        scal[1] = (reg_num > 0.0f) ? 1.0f : 0.0f;            // has_reg
    }
}

// ---------------- K3: masked CE sums ----------------
__global__ __launch_bounds__(THREADS) void k3_mask(
    const float* __restrict__ ce_buf, const unsigned char* __restrict__ flags,
    const float* __restrict__ scal, float* __restrict__ p3) {
    __shared__ float red[THREADS];
    const bool cond = scal[0] > 0.5f;
    float s_ce = 0.0f, s_cnt = 0.0f;
    for (int i = blockIdx.x * THREADS + threadIdx.x; i < BN;
         i += NBLK3 * THREADS) {
        unsigned f = flags[i];
        bool pos = f & 1, ign = f & 2, osp = f & 4, neg = f & 8;
        bool ospf = osp && !ign && (!cond || neg);
        if (pos || ospf) { s_ce += ce_buf[i]; s_cnt += 1.0f; }
    }
    float t1 = blk_reduce(s_ce,  red);
    float t2 = blk_reduce(s_cnt, red);
    if (threadIdx.x == 0) {
        p3[blockIdx.x]         = t1;
        p3[NBLK3 + blockIdx.x] = t2;
    }
}

// ---------------- K4: conf_loss (one wave, WMMA reductions) ----------------
__global__ __launch_bounds__(32) void k4_final(
    const float* __restrict__ p3, const float* __restrict__ scal,
    float* __restrict__ out) {
    const int lane = threadIdx.x;
    float sum_ce = wmma_sum(p3,          NBLK3 / 64, lane);
    float cnt    = wmma_sum(p3 + NBLK3,  NBLK3 / 64, lane);
    if (lane == 0) {
        float conf = sum_ce / fmaxf(cnt, 1.0f);
        out[2] = (scal[1] > 0.5f) ? conf : 0.0f;
    }
}

// ---------------- launch ----------------
extern "C" void kernel_launch(void* const* d_in, const int* in_sizes, int n_in,
                              void* d_out, int out_size, void* d_ws,
                              size_t ws_size, hipStream_t stream) {
    (void)in_sizes; (void)n_in; (void)out_size; (void)ws_size;
    const float* loc1 = (const float*)d_in[0];
    const float* loc2 = (const float*)d_in[1];
    const float* loct = (const float*)d_in[2];
    const float* conf = (const float*)d_in[3];
    const int*   ctg  = (const int*)d_in[4];
    const float* osp  = (const float*)d_in[5];
    const int*   ost  = (const int*)d_in[6];
    float*       out  = (float*)d_out;

    // workspace layout (~10 MB, L2-resident on MI455X's 192 MB L2)
    float*         ws      = (float*)d_ws;
    float*         ce_buf  = ws;                                  // BN f32
    unsigned*      key_buf = (unsigned*)(ws + BN);                // BN u32
    unsigned char* flags   = (unsigned char*)(ws + 2 * BN);       // BN u8
    float*         part    = ws + 2 * BN + BN / 4;                // 6*NBLK0
    float*         scal    = part + 6 * NBLK0;                    // 4
    float*         p3      = scal + 4;                            // 2*NBLK3

    k0_elem<<<NBLK0, THREADS, 0, stream>>>(loc1, loc2, loct, conf, ctg, osp,
                                           ost, ce_buf, key_buf, flags, part);
    k1_mine<<<BATCH, THREADS, 0, stream>>>(key_buf, flags);
    k2_scalars<<<1, 32, 0, stream>>>(part, scal, out);
    k3_mask<<<NBLK3, THREADS, 0, stream>>>(ce_buf, flags, scal, p3);
    k4_final<<<1, 32, 0, stream>>>(p3, scal, out);
}